// ROIPoolingLayer_48670569398569
// MI455X (gfx1250) — compile-verified
//
#include <hip/hip_runtime.h>
#include <math.h>

// ---------------------------------------------------------------------------
// ROI pooling via integral image, CDNA5 (gfx1250, wave32).
//
//   K1 rowscan_wmma : row-wise inclusive prefix sum of fm[c] using
//                     V_WMMA_F32_16X16X4_F32 (prefix = X * upper-tri ones,
//                     K=16 split into 4 chained K=4 WMMAs), carry broadcast
//                     via ds_bpermute_b32.
//   K3 colscan      : column cumsum + transpose to channel-innermost padded
//                     integral image ii[201][201][256] (LDS tile transpose).
//   K0 zero_borders : zero pad row r=0 and pad col k=0 (every launch).
//   K2 roi_gather   : per box: 49 bins x 4 corner lookups, each a contiguous
//                     256-channel (1 KB) coalesced load; LDS-staged transpose
//                     so output stores are coalesced.
//
// Workspace usage: rowcum 256*200*200 f32 (40.96 MB) followed by
//                  ii 201*201*256 f32 (41.37 MB)  => ~82.3 MB of d_ws.
// ---------------------------------------------------------------------------

typedef __attribute__((ext_vector_type(2))) float v2f;
typedef __attribute__((ext_vector_type(8))) float v8f;

#define C_   256
#define H_   200
#define W_   200
#define HP   201
#define WP   201
#define OH_  7
#define OW_  7
#define NBIN (OH_ * OW_)

// ---------------------------------------------------------------------------
// K1: row-wise inclusive prefix sum with f32 WMMA.
// grid = 256 channels * 13 strips of 16 rows, block = 1 wave (32 threads).
// A-matrix layout (16x4 f32): lanes 0-15 hold M=lane {K=4q+0,4q+1},
// lanes 16-31 hold M=lane-16 {K=4q+2,4q+3}. B mirrored. D: VGPR p holds
// row M=p (lanes 0-15, N=lane) / M=p+8 (lanes 16-31, N=lane-16).
// Each lane loads exactly the two floats it contributes per K-chunk
// (float2 at col = j0 + 4q + 2*hi) -> no register-array extracts.
// ---------------------------------------------------------------------------
__global__ __launch_bounds__(32) void rowscan_wmma(const float* __restrict__ F,
                                                   float* __restrict__ R) {
    const int strip = blockIdx.x % 13;
    const int c     = blockIdx.x / 13;
    const int lane  = threadIdx.x;
    const int n     = lane & 15;     // N index / A-row index for this lane
    const int hi    = lane >> 4;     // 0 = low half, 1 = high half
    const int r0    = strip * 16;
    const int rowA  = r0 + n;        // row this lane supplies for the A tile
    const bool rowok = (rowA < H_);

    const float* __restrict__ Fc = F + (size_t)c * (H_ * W_);
    float* __restrict__ Rc       = R + (size_t)c * (H_ * W_);
    const float* __restrict__ rowp = Fc + (size_t)(rowok ? rowA : 0) * W_;

    // Constant upper-triangular B fragments: U[k][j] = (k <= j), tile-local.
    v2f b[4];
#pragma unroll
    for (int q = 0; q < 4; ++q) {
        const int k0 = 4 * q + 2 * hi;
        b[q].x = (k0     <= n) ? 1.0f : 0.0f;
        b[q].y = (k0 + 1 <= n) ? 1.0f : 0.0f;
    }

    float carry[8];
#pragma unroll
    for (int p = 0; p < 8; ++p) carry[p] = 0.0f;

    // Byte address of the carry source lane for ds_bpermute:
    // lanes 0-15 read lane 15, lanes 16-31 read lane 31.
    const int permaddr = ((lane & 16) | 15) << 2;

    for (int t = 0; t < 13; ++t) {
        const int j0 = t * 16;

        // Speculative prefetch of the next tile (gfx1250 global_prefetch_b8).
        __builtin_prefetch(rowp + j0 + 16, 0, 0);

        // A fragments: per lane a direct 8B-aligned float2 at its own
        // (row, col) position; zero outside the 200x200 plane. The float2
        // never straddles the W=200 edge (even row stride, even chunk base).
        v2f a[4];
#pragma unroll
        for (int q = 0; q < 4; ++q) {
            const int col = j0 + 4 * q + 2 * hi;
            if (rowok && col < W_) {
                a[q].x = rowp[col];
                a[q].y = rowp[col + 1];
            } else {
                a[q].x = 0.0f;
                a[q].y = 0.0f;
            }
        }

        // Seed accumulator with running carry (same value for every column).
        v8f acc;
#pragma unroll
        for (int p = 0; p < 8; ++p) acc[p] = carry[p];

        // D = carry + X * U  (K=16 as 4 chained K=4 WMMAs).
        acc = __builtin_amdgcn_wmma_f32_16x16x4_f32(false, a[0], false, b[0],
                                                    (short)0, acc, false, false);
        acc = __builtin_amdgcn_wmma_f32_16x16x4_f32(false, a[1], false, b[1],
                                                    (short)0, acc, false, false);
        acc = __builtin_amdgcn_wmma_f32_16x16x4_f32(false, a[2], false, b[2],
                                                    (short)0, acc, false, false);
        acc = __builtin_amdgcn_wmma_f32_16x16x4_f32(false, a[3], false, b[3],
                                                    (short)0, acc, false, false);

        // Store D (guarded for the ragged last row-strip / col-tile).
        // Lanes 0-15 store 64B-contiguous runs per p; lanes 16-31 likewise.
#pragma unroll
        for (int p = 0; p < 8; ++p) {
            const int row = r0 + p + 8 * hi;
            const int col = j0 + n;
            if (row < H_ && col < W_) Rc[(size_t)row * W_ + col] = acc[p];
        }

        // New carry = D[:,15] (already includes the old carry), broadcast
        // from lane 15 / lane 31 with a native ds_bpermute_b32.
#pragma unroll
        for (int p = 0; p < 8; ++p) {
            carry[p] = __int_as_float(
                __builtin_amdgcn_ds_bpermute(permaddr, __float_as_int(acc[p])));
        }
    }
}

// ---------------------------------------------------------------------------
// K3: column cumsum of rowcum, transposed into channel-innermost padded
// integral image ii[r][k][c] with r,k shifted by +1.
// grid = 7 k-tiles * 8 c-tiles (32x32 each), block = 256 threads.
// ---------------------------------------------------------------------------
__global__ __launch_bounds__(256) void colscan_transpose(const float* __restrict__ R,
                                                         float* __restrict__ II) {
    __shared__ float smem[32 * 33];

    const int kt = blockIdx.x % 7;
    const int ct = blockIdx.x / 7;
    const int k0 = kt * 32;
    const int c0 = ct * 32;
    const int t  = threadIdx.x;
    const int l5 = t & 31;
    const int g  = t >> 5;  // 0..7

    float carry[4] = {0.0f, 0.0f, 0.0f, 0.0f};

    for (int r = 0; r < H_; ++r) {
        // Read phase: l5 = k (coalesced 128B runs), g+8s = channel.
#pragma unroll
        for (int s = 0; s < 4; ++s) {
            const int cc = g + 8 * s;
            const int kk = l5;
            float v = (k0 + kk < W_)
                          ? R[(size_t)(c0 + cc) * (H_ * W_) + (size_t)r * W_ + (k0 + kk)]
                          : 0.0f;
            carry[s] += v;
            smem[cc * 33 + kk] = carry[s];
        }
        __syncthreads();
        // Write phase: l5 = channel (coalesced 128B runs), g+8s = k.
#pragma unroll
        for (int s = 0; s < 4; ++s) {
            const int kw = g + 8 * s;
            const int cw = l5;
            if (k0 + kw < W_) {
                II[((size_t)(r + 1) * WP + (k0 + kw + 1)) * C_ + (c0 + cw)] =
                    smem[cw * 33 + kw];
            }
        }
        __syncthreads();
    }
}

// ---------------------------------------------------------------------------
// K0: zero the integral-image pad borders (row r=0 for all k, col k=0 for
// r=1..200). Must run every launch (workspace is not re-initialized).
// ---------------------------------------------------------------------------
__global__ void zero_borders(float* __restrict__ II) {
    const int idx = blockIdx.x * blockDim.x + threadIdx.x;
    const int n0  = WP * C_;        // r = 0 plane, contiguous at offset 0
    const int n1  = (HP - 1) * C_;  // k = 0 column, r = 1..200
    if (idx < n0) {
        II[idx] = 0.0f;
    } else if (idx < n0 + n1) {
        const int j = idx - n0;
        const int r = j / C_ + 1;
        const int c = j % C_;
        II[(size_t)r * WP * C_ + c] = 0.0f;
    }
}

// ---------------------------------------------------------------------------
// K2: per-box gather. block = box, thread = channel. Each corner lookup is a
// contiguous 256-channel (1 KB) load since ii is channel-innermost. Results
// staged through LDS so the final global store is fully coalesced.
// ---------------------------------------------------------------------------
__global__ __launch_bounds__(256) void roi_gather(const float* __restrict__ II,
                                                  const float* __restrict__ boxes,
                                                  float* __restrict__ out) {
    __shared__ float smem[C_ * NBIN];  // 12544 f32 = 49 KB (<< 320 KB/WGP)

    const int nbox = blockIdx.x;
    const int c    = threadIdx.x;

    // Box -> feature-map bin edges, exactly mirroring the reference math.
    const float sx = 0.25f;  // W/IMG_W = 200/800 (exact power of two)
    const float sy = 0.25f;  // H/IMG_H
    int x1 = (int)floorf(boxes[nbox * 4 + 0] * sx);
    int y1 = (int)floorf(boxes[nbox * 4 + 1] * sy);
    int x2 = (int)floorf(boxes[nbox * 4 + 2] * sx) + 1;
    int y2 = (int)floorf(boxes[nbox * 4 + 3] * sy) + 1;
    x1 = min(max(x1, 0), W_ - 1);
    y1 = min(max(y1, 0), H_ - 1);
    x2 = min(max(x2, x1 + 1), W_);
    y2 = min(max(y2, y1 + 1), H_);
    const int rh = y2 - y1;
    const int rw = x2 - x1;

#pragma unroll
    for (int bin = 0; bin < NBIN; ++bin) {
        const int i  = bin / OW_;
        const int j  = bin % OW_;
        const int r0 = y1 + (i * rh) / OH_;
        const int r1 = y1 + ((i + 1) * rh + OH_ - 1) / OH_;
        const int cl = x1 + (j * rw) / OW_;
        const int cr = x1 + ((j + 1) * rw + OW_ - 1) / OW_;

        const float s11 = II[((size_t)r1 * WP + cr) * C_ + c];
        const float s01 = II[((size_t)r0 * WP + cr) * C_ + c];
        const float s10 = II[((size_t)r1 * WP + cl) * C_ + c];
        const float s00 = II[((size_t)r0 * WP + cl) * C_ + c];

        const float total = s11 - s01 - s10 + s00;
        const float area  = (float)((r1 - r0) * (cr - cl));
        // stride 49 mod 64 banks -> conflict-free LDS writes
        smem[c * NBIN + bin] = total / area;
    }
    __syncthreads();

    float* __restrict__ outn = out + (size_t)nbox * (C_ * NBIN);
#pragma unroll
    for (int t = 0; t < NBIN; ++t) {
        outn[t * C_ + c] = smem[t * C_ + c];
    }
}

// ---------------------------------------------------------------------------
extern "C" void kernel_launch(void* const* d_in, const int* in_sizes, int n_in,
                              void* d_out, int out_size, void* d_ws, size_t ws_size,
                              hipStream_t stream) {
    const float* fm    = (const float*)d_in[0];  // [1,256,200,200] f32
    const float* boxes = (const float*)d_in[1];  // [1024,4] f32
    float* out         = (float*)d_out;          // [1024,256,7,7] f32

    float* rowcum = (float*)d_ws;                       // 256*200*200 f32
    float* ii     = rowcum + (size_t)C_ * H_ * W_;      // 201*201*256 f32

    const int nborder = WP * C_ + (HP - 1) * C_;
    zero_borders<<<(nborder + 255) / 256, 256, 0, stream>>>(ii);
    rowscan_wmma<<<C_ * 13, 32, 0, stream>>>(fm, rowcum);
    colscan_transpose<<<7 * 8, 256, 0, stream>>>(rowcum, ii);
    roi_gather<<<1024, 256, 0, stream>>>(ii, boxes, out);
}